// SoftmaxSquaredLoss_56934086476275
// MI455X (gfx1250) — compile-verified
//
#include <hip/hip_runtime.h>

// CDNA5 / gfx1250: wave32, WMMA 16x16x4 f32 as a rank-2 outer-product engine.
// hess = a*u^T + u*(-v)^T + diag(v), u=s, v=s^2, a=s*(q-s), q=sum(s^2).
typedef __attribute__((ext_vector_type(2))) float v2f;
typedef __attribute__((ext_vector_type(8))) float v8f;

#define C_DIM 256

__global__ __launch_bounds__(256) void softmax_loss_hess_kernel(
    const float* __restrict__ x, const float* __restrict__ y,
    float* __restrict__ loss, float* __restrict__ grad,
    float* __restrict__ hess)
{
    __shared__ float2 au_lds[C_DIM];   // (a, u)  -> A operand pair
    __shared__ float2 uv_lds[C_DIM];   // (u, v)  -> B operand pair (+diag v)
    __shared__ float  scr[5][8];       // cross-wave scratch: max, esum, l2, dot, q

    const int t    = threadIdx.x;
    const int b    = blockIdx.x;
    const int wave = t >> 5;
    const int lane = t & 31;

    const float* xrow = x + (size_t)b * C_DIM;
    const float* yrow = y + (size_t)b * C_DIM;

    // ---------------- Phase 1: softmax + fused reductions (W == 1.0) ----------
    float xv = xrow[t];

    // wave-level max, then cross-wave via LDS
    float wm = xv;
    #pragma unroll
    for (int o = 16; o > 0; o >>= 1) wm = fmaxf(wm, __shfl_xor(wm, o, 32));
    if (lane == 0) scr[0][wave] = wm;
    __syncthreads();
    float m = scr[0][0];
    #pragma unroll
    for (int w = 1; w < 8; ++w) m = fmaxf(m, scr[0][w]);

    float e  = __expf(xv - m);
    float ws = e;
    #pragma unroll
    for (int o = 16; o > 0; o >>= 1) ws += __shfl_xor(ws, o, 32);
    if (lane == 0) scr[1][wave] = ws;
    __syncthreads();
    float esum = 0.0f;
    #pragma unroll
    for (int w = 0; w < 8; ++w) esum += scr[1][w];

    const float s   = e * (1.0f / esum);
    const float dif = s - yrow[t];

    // fused three sums: l2 = sum(diff^2), dot = sum(s*diff), q = sum(s^2)
    float r0 = dif * dif, r1 = s * dif, r2 = s * s;
    #pragma unroll
    for (int o = 16; o > 0; o >>= 1) {
        r0 += __shfl_xor(r0, o, 32);
        r1 += __shfl_xor(r1, o, 32);
        r2 += __shfl_xor(r2, o, 32);
    }
    if (lane == 0) { scr[2][wave] = r0; scr[3][wave] = r1; scr[4][wave] = r2; }
    __syncthreads();
    float l2 = 0.0f, dot = 0.0f, q = 0.0f;
    #pragma unroll
    for (int w = 0; w < 8; ++w) { l2 += scr[2][w]; dot += scr[3][w]; q += scr[4][w]; }

    // small outputs
    grad[(size_t)b * C_DIM + t] = s * (dif - dot);
    if (t == 0) loss[b] = 0.5f * l2;

    // stage rank-2 factors (packed pairs -> single ds_load_b64 per operand)
    au_lds[t] = make_float2(s * (q - s), s);   // (a, u)
    uv_lds[t] = make_float2(s, s * s);         // (u, v)
    __syncthreads();

    // ---------------- Phase 2: hessian tiles via V_WMMA_F32_16X16X4_F32 ------
    // 16x16 grid of 16x16 tiles; 8 waves x 2 tile-rows x 16 tile-cols.
    const int half = lane >> 4;   // 0: lanes 0-15 carry K0/K1, 1: lanes 16-31 carry K2/K3 (=0)
    const int l15  = lane & 15;
    const float zsel = half ? 0.0f : 1.0f;     // not used for loads; loads are unconditional
    (void)zsel;
    float* hrow = hess + ((size_t)b << 16);    // b * 256 * 256

    #pragma unroll
    for (int tr = 0; tr < 2; ++tr) {
        const int ti = wave * 2 + tr;

        // A (16x4 f32): lanes 0-15 -> M=lane, VGPR0=K0 col (a), VGPR1=K1 col (u);
        // lanes 16-31 -> zero K2/K3 cols. Unconditional load, then cndmask.
        const float2 Ad = au_lds[ti * 16 + l15];
        v2f A;
        A.x = half ? 0.0f : Ad.x;
        A.y = half ? 0.0f : Ad.y;

        // base pointer for this lane's store column in tile-row ti
        float* prow = hrow + (size_t)(ti * 16 + half * 8) * C_DIM + l15;

        for (int tj = 0; tj < 16; ++tj) {
            // B (4x16 f32): lanes 0-15 -> N=lane, VGPR0=row K0 (u), VGPR1=row K1 (-v).
            const float2 Bd = uv_lds[tj * 16 + l15];
            v2f Bv;
            Bv.x = half ? 0.0f : Bd.x;
            Bv.y = half ? 0.0f : -Bd.y;

            v8f c = {};
            // (neg_a, A, neg_b, B, c_mod, C, reuse_a, reuse_b)
            c = __builtin_amdgcn_wmma_f32_16x16x4_f32(
                    false, A, false, Bv, (short)0, c, false, false);

            // diagonal correction: +v on i==j. Bd.y is exactly v[tj*16+l15],
            // which is the needed v[ti*16+l15] when ti==tj -> no extra LDS load.
            if (ti == tj) {
                #pragma unroll
                for (int k = 0; k < 8; ++k) {
                    const int mrow = k + half * 8;   // row-in-tile held by VGPR k
                    c[k] += (mrow == l15) ? Bd.y : 0.0f;
                }
            }

            // stores: VGPR k -> row (ti*16 + k + half*8), 16 contiguous floats per
            // half-wave; k*256 folds into the global_store immediate offset.
            float* p = prow + tj * 16;
            #pragma unroll
            for (int k = 0; k < 8; ++k) {
                __builtin_nontemporal_store(c[k], p + (size_t)k * C_DIM);
            }
        }
    }
}

extern "C" void kernel_launch(void* const* d_in, const int* in_sizes, int n_in,
                              void* d_out, int out_size, void* d_ws, size_t ws_size,
                              hipStream_t stream) {
    const float* x = (const float*)d_in[0];
    const float* y = (const float*)d_in[1];
    const int B = in_sizes[0] / C_DIM;            // 2048

    float* out  = (float*)d_out;
    float* loss = out;                            // [B,1]
    float* grad = out + B;                        // [B,C]
    float* hess = grad + (size_t)B * C_DIM;       // [B,C,C]

    softmax_loss_hess_kernel<<<dim3(B), dim3(256), 0, stream>>>(x, y, loss, grad, hess);
}